// TauAttentionGNN_6176162972390
// MI455X (gfx1250) — compile-verified
//
#include <hip/hip_runtime.h>
#include <hip/hip_bf16.h>
#include <math.h>

// ---------------------------------------------------------------------------
// TauAttentionGNN for MI455X (gfx1250, wave32).
// Dense GEMMs use V_WMMA_F32_16X16X4_F32 (full fp32, matches reference).
// Edge phase (the real bottleneck) uses coalesced float4 gathers out of L2
// (h fits in 192MB L2) and hardware global_atomic_add_f32 scatter.
// ---------------------------------------------------------------------------

#define NNODES  100000
#define NEDGES  1000000
#define IN_DIM  128
#define HID     64
#define OUT_DIM 64
#define TAU_DIM 8
#define MAX_REC 5

// padded LDS row strides (stride % 64 == 4 -> conflict-free b64 A-operand reads)
#define XS_LD   (IN_DIM + 4)   // 132
#define HS_LD   (HID + 4)      // 68

typedef __attribute__((ext_vector_type(2))) float v2f;
typedef __attribute__((ext_vector_type(8))) float v8f;

// D = A(16x4) * B(4x16) + C, fp32, wave32
static __device__ __forceinline__ v8f wmma_f32(v2f a, v2f b, v8f c) {
  return __builtin_amdgcn_wmma_f32_16x16x4_f32(false, a, false, b,
                                               (short)0, c, false, false);
}

static __device__ __forceinline__ float sigmoidf_(float x) {
  return 1.0f / (1.0f + __expf(-x));
}

// ---------------------------------------------------------------------------
// init: h = relu(x @ W_in^T + b_in); tau = mean(sigmoid(relu(x@W_t1^T+b_t1)@W_t2^T+b_t2))
// grid: NNODES/16 blocks of 128 threads (4 waves). Wave w -> output cols [16w,16w+16)
// ---------------------------------------------------------------------------
__global__ __launch_bounds__(128) void init_kernel(
    const float* __restrict__ x,
    const float* __restrict__ W_in, const float* __restrict__ b_in,
    const float* __restrict__ W_t1, const float* __restrict__ b_t1,
    const float* __restrict__ W_t2, const float* __restrict__ b_t2,
    float* __restrict__ h, float* __restrict__ tau)
{
  __shared__ float xs[16 * XS_LD];   // x tile, padded
  __shared__ float t1s[16 * HS_LD];  // relu(x@W_t1^T+b_t1) tile, padded

  const int node0 = blockIdx.x * 16;
  const int tid   = threadIdx.x;

  // stage x tile [16 x 128] into LDS (coalesced global reads)
  for (int i = tid; i < 16 * IN_DIM; i += 128) {
    int r = i >> 7, c = i & (IN_DIM - 1);
    xs[r * XS_LD + c] = x[(size_t)(node0 + r) * IN_DIM + c];
  }
  __syncthreads();

  const int lane = tid & 31;
  const int wave = tid >> 5;
  const int hs   = lane >> 4;   // half-wave select
  const int lm   = lane & 15;
  const int n0   = wave * 16;   // output-column tile base

  v8f accH = {}; v8f accT = {};
  for (int k = 0; k < IN_DIM; k += 4) {
    const int ka = k + 2 * hs;
    v2f a  = *(const v2f*)&xs[lm * XS_LD + ka];
    v2f bH = *(const v2f*)&W_in[(size_t)(n0 + lm) * IN_DIM + ka];
    v2f bT = *(const v2f*)&W_t1[(size_t)(n0 + lm) * IN_DIM + ka];
    accH = wmma_f32(a, bH, accH);   // shared A operand across the two WMMAs
    accT = wmma_f32(a, bT, accT);
  }

  const float bh = b_in[n0 + lm];
  const float bt = b_t1[n0 + lm];
  #pragma unroll
  for (int v = 0; v < 8; ++v) {
    const int row = v + 8 * hs;                       // M within tile
    float hv = accH[v] + bh;  hv = hv > 0.f ? hv : 0.f;
    h[(size_t)(node0 + row) * HID + n0 + lm] = hv;
    float tv = accT[v] + bt;  tv = tv > 0.f ? tv : 0.f;
    t1s[row * HS_LD + n0 + lm] = tv;
  }
  __syncthreads();

  // tau: 16 threads, one node row each (8 dot-products of length 64 -> sigmoid -> mean)
  if (tid < 16) {
    float s = 0.f;
    #pragma unroll
    for (int t = 0; t < TAU_DIM; ++t) {
      float d = b_t2[t];
      for (int k = 0; k < HID; ++k) d += t1s[tid * HS_LD + k] * W_t2[t * HID + k];
      s += sigmoidf_(d);
    }
    tau[node0 + tid] = s * (1.0f / TAU_DIM);
  }
}

// ---------------------------------------------------------------------------
// zero the segment-sum accumulator
// ---------------------------------------------------------------------------
__global__ __launch_bounds__(256) void zero_kernel(float4* __restrict__ agg, int n4) {
  int i = blockIdx.x * 256 + threadIdx.x;
  if (i < n4) { float4 z = {0.f, 0.f, 0.f, 0.f}; agg[i] = z; }
}

// ---------------------------------------------------------------------------
// edge: agg[row] += |h[row]-h[col]| * tau[row]   (16 threads per edge, float4)
// L2-resident gathers + hardware fp32 atomics (no-return form).
// ---------------------------------------------------------------------------
__global__ __launch_bounds__(256) void edge_kernel(
    const long long* __restrict__ ei,   // [2, E] int64
    const float* __restrict__ h, const float* __restrict__ tau,
    float* __restrict__ agg)
{
  const int gid = blockIdx.x * 256 + threadIdx.x;
  const int e   = gid >> 4;
  const int c4  = (gid & 15) * 4;
  if (e >= NEDGES) return;

  const long long r = ei[e];
  const long long c = ei[NEDGES + e];
  const float t = tau[r];

  const float4 hr = *(const float4*)&h[(size_t)r * HID + c4];
  const float4 hc = *(const float4*)&h[(size_t)c * HID + c4];

  float4 d;
  d.x = fabsf(hr.x - hc.x) * t;
  d.y = fabsf(hr.y - hc.y) * t;
  d.z = fabsf(hr.z - hc.z) * t;
  d.w = fabsf(hr.w - hc.w) * t;

  float* ap = &agg[(size_t)r * HID + c4];
  __hip_atomic_fetch_add(ap + 0, d.x, __ATOMIC_RELAXED, __HIP_MEMORY_SCOPE_AGENT);
  __hip_atomic_fetch_add(ap + 1, d.y, __ATOMIC_RELAXED, __HIP_MEMORY_SCOPE_AGENT);
  __hip_atomic_fetch_add(ap + 2, d.z, __ATOMIC_RELAXED, __HIP_MEMORY_SCOPE_AGENT);
  __hip_atomic_fetch_add(ap + 3, d.w, __ATOMIC_RELAXED, __HIP_MEMORY_SCOPE_AGENT);
}

// ---------------------------------------------------------------------------
// GRU cell: h = GRUCell(agg, h). One block per 16-node tile, 4 waves; wave w
// owns hidden cols [16w,16w+16) and keeps 6 WMMA accumulators (r/z/n x ih/hh).
// In-place h update is safe: each block owns its rows; cross-block ordering by
// kernel boundary.
// ---------------------------------------------------------------------------
__global__ __launch_bounds__(128) void gru_kernel(
    const float* __restrict__ agg, float* __restrict__ h,
    const float* __restrict__ W_ih, const float* __restrict__ b_ih,
    const float* __restrict__ W_hh, const float* __restrict__ b_hh)
{
  __shared__ float as[16 * HS_LD];
  __shared__ float hl[16 * HS_LD];

  const int node0 = blockIdx.x * 16;
  const int tid   = threadIdx.x;

  for (int i = tid; i < 16 * HID; i += 128) {
    int r = i >> 6, c = i & (HID - 1);
    as[r * HS_LD + c] = agg[(size_t)(node0 + r) * HID + c];
    hl[r * HS_LD + c] = h[(size_t)(node0 + r) * HID + c];
  }
  __syncthreads();

  const int lane = tid & 31;
  const int wave = tid >> 5;
  const int hs   = lane >> 4;
  const int lm   = lane & 15;
  const int c0   = wave * 16;

  v8f ir = {}, iz = {}, inn = {}, hr = {}, hz = {}, hn = {};
  for (int k = 0; k < HID; k += 4) {
    const int ka = k + 2 * hs;
    v2f aA = *(const v2f*)&as[lm * HS_LD + ka];
    v2f aH = *(const v2f*)&hl[lm * HS_LD + ka];
    v2f br = *(const v2f*)&W_ih[(size_t)(      c0 + lm) * HID + ka];
    v2f bz = *(const v2f*)&W_ih[(size_t)( 64 + c0 + lm) * HID + ka];
    v2f bn = *(const v2f*)&W_ih[(size_t)(128 + c0 + lm) * HID + ka];
    v2f cr = *(const v2f*)&W_hh[(size_t)(      c0 + lm) * HID + ka];
    v2f cz = *(const v2f*)&W_hh[(size_t)( 64 + c0 + lm) * HID + ka];
    v2f cn = *(const v2f*)&W_hh[(size_t)(128 + c0 + lm) * HID + ka];
    ir  = wmma_f32(aA, br, ir);
    iz  = wmma_f32(aA, bz, iz);
    inn = wmma_f32(aA, bn, inn);
    hr  = wmma_f32(aH, cr, hr);
    hz  = wmma_f32(aH, cz, hz);
    hn  = wmma_f32(aH, cn, hn);
  }

  const float bir = b_ih[      c0 + lm], bhr = b_hh[      c0 + lm];
  const float biz = b_ih[ 64 + c0 + lm], bhz = b_hh[ 64 + c0 + lm];
  const float bin = b_ih[128 + c0 + lm], bhn = b_hh[128 + c0 + lm];

  #pragma unroll
  for (int v = 0; v < 8; ++v) {
    const int row = v + 8 * hs;
    const float r = sigmoidf_((ir[v] + bir) + (hr[v] + bhr));
    const float z = sigmoidf_((iz[v] + biz) + (hz[v] + bhz));
    const float n = tanhf((inn[v] + bin) + r * (hn[v] + bhn));
    const float hv = hl[row * HS_LD + c0 + lm];
    h[(size_t)(node0 + row) * HID + c0 + lm] = (1.0f - z) * n + z * hv;
  }
}

// ---------------------------------------------------------------------------
// out = h @ W_out^T + b_out
// ---------------------------------------------------------------------------
__global__ __launch_bounds__(128) void out_kernel(
    const float* __restrict__ h,
    const float* __restrict__ W_out, const float* __restrict__ b_out,
    float* __restrict__ out)
{
  __shared__ float hl[16 * HS_LD];

  const int node0 = blockIdx.x * 16;
  const int tid   = threadIdx.x;

  for (int i = tid; i < 16 * HID; i += 128) {
    int r = i >> 6, c = i & (HID - 1);
    hl[r * HS_LD + c] = h[(size_t)(node0 + r) * HID + c];
  }
  __syncthreads();

  const int lane = tid & 31;
  const int wave = tid >> 5;
  const int hs   = lane >> 4;
  const int lm   = lane & 15;
  const int n0   = wave * 16;

  v8f acc = {};
  for (int k = 0; k < HID; k += 4) {
    const int ka = k + 2 * hs;
    v2f a = *(const v2f*)&hl[lm * HS_LD + ka];
    v2f b = *(const v2f*)&W_out[(size_t)(n0 + lm) * HID + ka];
    acc = wmma_f32(a, b, acc);
  }

  const float bb = b_out[n0 + lm];
  #pragma unroll
  for (int v = 0; v < 8; ++v) {
    const int row = v + 8 * hs;
    out[(size_t)(node0 + row) * OUT_DIM + n0 + lm] = acc[v] + bb;
  }
}

// ---------------------------------------------------------------------------
extern "C" void kernel_launch(void* const* d_in, const int* in_sizes, int n_in,
                              void* d_out, int out_size, void* d_ws, size_t ws_size,
                              hipStream_t stream) {
  const float*     x     = (const float*)d_in[0];
  const long long* ei    = (const long long*)d_in[1];
  const float*     W_in  = (const float*)d_in[2];
  const float*     b_in  = (const float*)d_in[3];
  const float*     W_t1  = (const float*)d_in[4];
  const float*     b_t1  = (const float*)d_in[5];
  const float*     W_t2  = (const float*)d_in[6];
  const float*     b_t2  = (const float*)d_in[7];
  const float*     W_ih  = (const float*)d_in[8];
  const float*     b_ih  = (const float*)d_in[9];
  const float*     W_hh  = (const float*)d_in[10];
  const float*     b_hh  = (const float*)d_in[11];
  const float*     W_out = (const float*)d_in[12];
  const float*     b_out = (const float*)d_in[13];
  float*           out   = (float*)d_out;

  // workspace: h [N*HID] | agg [N*HID] | tau [N]   (~51.6 MB)
  float* h   = (float*)d_ws;
  float* agg = h + (size_t)NNODES * HID;
  float* tau = agg + (size_t)NNODES * HID;

  const int tiles = NNODES / 16;                 // 6250
  const int n4    = NNODES * HID / 4;            // 1.6M float4

  init_kernel<<<tiles, 128, 0, stream>>>(x, W_in, b_in, W_t1, b_t1, W_t2, b_t2, h, tau);

  for (int it = 0; it < MAX_REC; ++it) {
    zero_kernel<<<(n4 + 255) / 256, 256, 0, stream>>>((float4*)agg, n4);
    edge_kernel<<<(NEDGES * 16) / 256, 256, 0, stream>>>(ei, h, tau, agg);
    gru_kernel<<<tiles, 128, 0, stream>>>(agg, h, W_ih, b_ih, W_hh, b_hh);
  }

  out_kernel<<<tiles, 128, 0, stream>>>(h, W_out, b_out, out);
}